// CoNystromAttention_87548613362291
// MI455X (gfx1250) — compile-verified
//
#include <hip/hip_runtime.h>
#include <math.h>

typedef __attribute__((ext_vector_type(2))) float v2f;
typedef __attribute__((ext_vector_type(8))) float v8f;

#define BATCH 4
#define NHEAD 16
#define SEQ   4096
#define HDIM  64
#define NLMK  64
#define EMB   1024
#define BH    (BATCH * NHEAD)

// Row stride (floats) for 64-wide LDS tiles: even (8B-aligned pairs) and
// 66 mod 64 == 2 -> 16 rows hit distinct even banks. 34 for 32-wide tiles.
#define ST64 66
#define ST32 34

// ---------------------------------------------------------------------------
// fp32 WMMA: D = A(16x4) @ B(4x16) + C(16x16).  Lane layout per ISA 7.12.2:
//   A: lanes 0-15 M=lane K={k0,k0+1}; lanes 16-31 M=lane-16 K={k0+2,k0+3}
//   B: vgpr j lanes 0-15 K=k0+j N=lane; lanes 16-31 K=k0+2+j
//   C: vgpr v lanes 0-15 M=v N=lane; lanes 16-31 M=v+8
// ---------------------------------------------------------------------------
__device__ __forceinline__ v8f wmma_f32(v2f a, v2f b, v8f c) {
  return __builtin_amdgcn_wmma_f32_16x16x4_f32(false, a, false, b, (short)0, c,
                                               false, false);
}

__device__ __forceinline__ v2f ldpair(const float* p) {
  return *(const v2f*)p;  // one ds_load_b64, 8B-aligned by construction
}

// C(64x64) = A @ B with A row-major [row][k] and B stored TRANSPOSED
// Bt[col][k] (so both operands are contiguous 8-byte pair loads).
// 8 waves, 2 tiles each.
__device__ __forceinline__ void mm64T(float (*Cm)[ST64], const float (*Am)[ST64],
                                      const float (*Bt)[ST64], int w, int lo,
                                      int hi) {
#pragma unroll
  for (int q = 0; q < 2; ++q) {
    const int ti = 2 * w + q, rt = ti >> 2, ct = ti & 3;
    v8f acc = {};
#pragma unroll
    for (int kk = 0; kk < 64; kk += 4) {
      v2f a = ldpair(&Am[rt * 16 + lo][kk + 2 * hi]);
      v2f b = ldpair(&Bt[ct * 16 + lo][kk + 2 * hi]);
      acc = wmma_f32(a, b, acc);
    }
#pragma unroll
    for (int v = 0; v < 8; ++v)
      Cm[rt * 16 + v + 8 * hi][ct * 16 + lo] = acc[v];
  }
}

// Variants with normal-orientation B (vertical pair fetch) for the cold
// Newton-Schulz kernel; optional fused  C = alpha*I - A@B  epilogue.
__device__ __forceinline__ v2f ldvert(const float (*Bm)[ST64], int k, int col) {
  v2f b;
  b.x = Bm[k][col];
  b.y = Bm[k + 1][col];
  return b;
}

__device__ __forceinline__ void mm64V(float (*Cm)[ST64], const float (*Am)[ST64],
                                      const float (*Bm)[ST64], int w, int lo,
                                      int hi) {
#pragma unroll
  for (int q = 0; q < 2; ++q) {
    const int ti = 2 * w + q, rt = ti >> 2, ct = ti & 3;
    v8f acc = {};
#pragma unroll
    for (int kk = 0; kk < 64; kk += 4) {
      v2f a = ldpair(&Am[rt * 16 + lo][kk + 2 * hi]);
      v2f b = ldvert(Bm, kk + 2 * hi, ct * 16 + lo);
      acc = wmma_f32(a, b, acc);
    }
#pragma unroll
    for (int v = 0; v < 8; ++v)
      Cm[rt * 16 + v + 8 * hi][ct * 16 + lo] = acc[v];
  }
}

__device__ __forceinline__ void mm64V_negI(float (*Cm)[ST64],
                                           const float (*Am)[ST64],
                                           const float (*Bm)[ST64], float alpha,
                                           int w, int lo, int hi) {
#pragma unroll
  for (int q = 0; q < 2; ++q) {
    const int ti = 2 * w + q, rt = ti >> 2, ct = ti & 3;
    v8f acc = {};
#pragma unroll
    for (int kk = 0; kk < 64; kk += 4) {
      v2f a = ldpair(&Am[rt * 16 + lo][kk + 2 * hi]);
      v2f b = ldvert(Bm, kk + 2 * hi, ct * 16 + lo);
      acc = wmma_f32(a, b, acc);
    }
#pragma unroll
    for (int v = 0; v < 8; ++v) {
      const int r = rt * 16 + v + 8 * hi, c = ct * 16 + lo;
      Cm[r][c] = ((r == c) ? alpha : 0.0f) - acc[v];
    }
  }
}

// ---------------------------------------------------------------------------
// K1: QKV[b,h,t,d] = sum_e X[b,e,t]*Wq[h,d,e] + bq[h,d]
// grid (16 heads, 32 t-tiles, 4 batch), block 256 (8 waves): 128 x 64 tile.
// ---------------------------------------------------------------------------
__global__ __launch_bounds__(256) void proj_kernel(
    const float* __restrict__ X, const float* __restrict__ Wq,
    const float* __restrict__ bq, float* __restrict__ QKV) {
  const int h = blockIdx.x;
  const int c0 = h * 64;
  const int t0 = blockIdx.y * 128;
  const int b = blockIdx.z;
  const int tid = threadIdx.x;
  const int w = tid >> 5, lane = tid & 31, lo = lane & 15, hi = lane >> 4;

  __shared__ float ldsA[128][ST32];  // A[t][e]  (k innermost)
  __shared__ float ldsBt[64][ST32];  // Bt[c][e] = W[c][e] (k innermost)

  v8f acc[4] = {};

  for (int e0 = 0; e0 < EMB; e0 += 32) {
    __syncthreads();
#pragma unroll
    for (int i = 0; i < 16; ++i) {
      const int idx = tid + i * 256;
      const int e = idx >> 7, t = idx & 127;  // t fast -> coalesced global
      ldsA[t][e] = X[(size_t)b * EMB * SEQ + (size_t)(e0 + e) * SEQ + t0 + t];
    }
#pragma unroll
    for (int i = 0; i < 8; ++i) {
      const int idx = tid + i * 256;
      const int c = idx >> 5, e = idx & 31;  // e fast -> coalesced global
      ldsBt[c][e] = Wq[(size_t)(c0 + c) * EMB + e0 + e];
    }
    __syncthreads();
#pragma unroll
    for (int kk = 0; kk < 32; kk += 4) {
      const v2f a = ldpair(&ldsA[w * 16 + lo][kk + 2 * hi]);
#pragma unroll
      for (int ct = 0; ct < 4; ++ct) {
        const v2f bb = ldpair(&ldsBt[ct * 16 + lo][kk + 2 * hi]);
        acc[ct] = wmma_f32(a, bb, acc[ct]);
      }
    }
  }

#pragma unroll
  for (int ct = 0; ct < 4; ++ct) {
    const int dd = ct * 16 + lo;
    const float bias = bq[h * HDIM + dd];
#pragma unroll
    for (int v = 0; v < 8; ++v) {
      const int t = t0 + w * 16 + v + 8 * hi;
      QKV[((size_t)(b * NHEAD + h) * SEQ + t) * HDIM + dd] = acc[ct][v] + bias;
    }
  }
}

// K2: landmark mean pooling
__global__ __launch_bounds__(256) void pool_kernel(const float* __restrict__ QKV,
                                                   float* __restrict__ Qt) {
  const int idx = blockIdx.x * 256 + threadIdx.x;  // over BH*64*64
  const int dd = idx & 63, l = (idx >> 6) & 63, bh = idx >> 12;
  const float* base = QKV + ((size_t)bh * SEQ + l * 64) * HDIM + dd;
  float s = 0.0f;
#pragma unroll 8
  for (int j = 0; j < 64; ++j) s += base[(size_t)j * HDIM];
  Qt[idx] = s * (1.0f / 64.0f);
}

// K3: GammaD = rownorm(exp(Qt@Qt^T/8)); global max colsum for the NS scale.
__global__ __launch_bounds__(256) void gamma_kernel(const float* __restrict__ Qt,
                                                    float* __restrict__ GammaD,
                                                    int* __restrict__ scaleMax) {
  const int bh = blockIdx.x, tid = threadIdx.x;
  const int w = tid >> 5, lane = tid & 31, lo = lane & 15, hi = lane >> 4;
  __shared__ float lq[64][ST64];
  __shared__ float ls[64][ST64];
  __shared__ float rs[64];
  __shared__ float cred[64];

#pragma unroll
  for (int i = 0; i < 16; ++i) {
    const int idx = tid + i * 256;
    lq[idx >> 6][idx & 63] = Qt[(size_t)bh * 4096 + idx];
  }
  __syncthreads();

  // S = Qt @ Qt^T : B operand is Qt^T, so Bt = Qt itself (row-major k) ✓
#pragma unroll
  for (int q = 0; q < 2; ++q) {
    const int ti = 2 * w + q, rt = ti >> 2, ct = ti & 3;
    v8f acc = {};
#pragma unroll
    for (int kk = 0; kk < 64; kk += 4) {
      const v2f a = ldpair(&lq[rt * 16 + lo][kk + 2 * hi]);
      const v2f b = ldpair(&lq[ct * 16 + lo][kk + 2 * hi]);
      acc = wmma_f32(a, b, acc);
    }
#pragma unroll
    for (int v = 0; v < 8; ++v)
      ls[rt * 16 + v + 8 * hi][ct * 16 + lo] = __expf(acc[v] * 0.125f);
  }
  __syncthreads();
  if (tid < 64) {
    float s = 0.0f;
    for (int j = 0; j < 64; ++j) s += ls[tid][j];
    rs[tid] = s;
  }
  __syncthreads();
#pragma unroll
  for (int i = 0; i < 16; ++i) {
    const int idx = tid + i * 256;
    const int r = idx >> 6, c = idx & 63;
    const float vv = ls[r][c] / rs[r];
    ls[r][c] = vv;
    GammaD[(size_t)bh * 4096 + idx] = vv;
  }
  __syncthreads();
  if (tid < 64) {
    float s = 0.0f;
    for (int r = 0; r < 64; ++r) s += ls[r][tid];
    cred[tid] = s;
  }
  __syncthreads();
  if (tid == 0) {
    float mx = 0.0f;
    for (int j = 0; j < 64; ++j) mx = fmaxf(mx, cred[j]);
    atomicMax(scaleMax, __float_as_int(mx));  // positive floats order as ints
  }
}

__global__ void init_kernel(int* scaleMax) { *scaleMax = 0; }

// K4: Newton-Schulz pseudo-inverse, 6 iters in LDS (dynamic).
__global__ __launch_bounds__(256) void newton_kernel(
    const float* __restrict__ GammaD, const int* __restrict__ scaleMax,
    float* __restrict__ GammaInv) {
  extern __shared__ float smem[];
  float(*Km)[ST64] = (float(*)[ST64])smem;
  float(*Vm)[ST64] = (float(*)[ST64])(smem + 64 * ST64);
  float(*KV)[ST64] = (float(*)[ST64])(smem + 2 * 64 * ST64);
  float(*T1)[ST64] = (float(*)[ST64])(smem + 3 * 64 * ST64);
  float(*T2)[ST64] = (float(*)[ST64])(smem + 4 * 64 * ST64);
  const int bh = blockIdx.x, tid = threadIdx.x;
  const int w = tid >> 5, lane = tid & 31, lo = lane & 15, hi = lane >> 4;

#pragma unroll
  for (int i = 0; i < 16; ++i) {
    const int idx = tid + i * 256;
    Km[idx >> 6][idx & 63] = GammaD[(size_t)bh * 4096 + idx];
  }
  __syncthreads();
  // rows of GammaD sum to 1 exactly -> scale = 1 / max colsum
  const float scale = 1.0f / __int_as_float(*scaleMax);
#pragma unroll
  for (int i = 0; i < 16; ++i) {
    const int idx = tid + i * 256;
    const int r = idx >> 6, c = idx & 63;
    Vm[r][c] = scale * Km[c][r];  // V0 = scale * K^T
  }
  __syncthreads();

  for (int it = 0; it < 6; ++it) {
    mm64V(KV, Km, Vm, w, lo, hi);
    __syncthreads();
#pragma unroll
    for (int i = 0; i < 16; ++i) {
      const int idx = tid + i * 256;
      const int r = idx >> 6, c = idx & 63;
      T1[r][c] = ((r == c) ? 7.0f : 0.0f) - KV[r][c];
    }
    __syncthreads();
    mm64V_negI(T2, KV, T1, 15.0f, w, lo, hi);  // T2 = 15I - KV@(7I-KV)
    __syncthreads();
    mm64V_negI(T1, KV, T2, 13.0f, w, lo, hi);  // T1 = 13I - KV@T2
    __syncthreads();
    mm64V(T2, Vm, T1, w, lo, hi);              // T2 = V @ T1
    __syncthreads();
#pragma unroll
    for (int i = 0; i < 16; ++i) {
      const int idx = tid + i * 256;
      Vm[idx >> 6][idx & 63] = 0.25f * T2[idx >> 6][idx & 63];
    }
    __syncthreads();
  }
#pragma unroll
  for (int i = 0; i < 16; ++i) {
    const int idx = tid + i * 256;
    GammaInv[(size_t)bh * 4096 + idx] = Vm[idx >> 6][idx & 63];
  }
}

// K5: streaming DV = rownorm(exp(Qt@K^T/8)) @ V, then M2 = GammaInv @ DV.
__global__ __launch_bounds__(256) void deltav_kernel(
    const float* __restrict__ QKV, const float* __restrict__ Qt,
    const float* __restrict__ GammaInv, float* __restrict__ M2) {
  extern __shared__ float smem[];
  float(*lq)[ST64] = (float(*)[ST64])smem;                    // Qt / GammaInv
  float(*lk)[ST64] = (float(*)[ST64])(smem + 64 * ST64);      // Kc[t][d]
  float(*lkT)[ST64] = (float(*)[ST64])(smem + 2 * 64 * ST64); // Vc^T[d][t]
  float(*lp)[ST64] = (float(*)[ST64])(smem + 3 * 64 * ST64);  // P / M2
  float* den = smem + 4 * 64 * ST64;                          // 64
  const int bh = blockIdx.x, tid = threadIdx.x;
  const int w = tid >> 5, lane = tid & 31, lo = lane & 15, hi = lane >> 4;

#pragma unroll
  for (int i = 0; i < 16; ++i) {
    const int idx = tid + i * 256;
    lq[idx >> 6][idx & 63] = Qt[(size_t)bh * 4096 + idx];
  }
  if (tid < 64) den[tid] = 0.0f;

  v8f accN[2] = {};
  for (int tc = 0; tc < SEQ / 64; ++tc) {
    __syncthreads();
#pragma unroll
    for (int i = 0; i < 16; ++i) {
      const int idx = tid + i * 256;
      const int t = idx >> 6, d = idx & 63;
      const float val = QKV[((size_t)bh * SEQ + tc * 64) * HDIM + idx];
      lk[t][d] = val;   // K^T as Bt (row-major k=d)
      lkT[d][t] = val;  // V as Bt   (row-major k=t)
    }
    __syncthreads();
    // P = exp(Qt @ Kc^T / 8):  Bt = lk
#pragma unroll
    for (int q = 0; q < 2; ++q) {
      const int ti = 2 * w + q, rt = ti >> 2, ct = ti & 3;
      v8f acc = {};
#pragma unroll
      for (int kk = 0; kk < 64; kk += 4) {
        const v2f a = ldpair(&lq[rt * 16 + lo][kk + 2 * hi]);
        const v2f b = ldpair(&lk[ct * 16 + lo][kk + 2 * hi]);
        acc = wmma_f32(a, b, acc);
      }
#pragma unroll
      for (int v = 0; v < 8; ++v)
        lp[rt * 16 + v + 8 * hi][ct * 16 + lo] = __expf(acc[v] * 0.125f);
    }
    __syncthreads();
    // NUM += P @ Vc:  Bt = lkT
#pragma unroll
    for (int q = 0; q < 2; ++q) {
      const int ti = 2 * w + q, rt = ti >> 2, ct = ti & 3;
#pragma unroll
      for (int kk = 0; kk < 64; kk += 4) {
        const v2f a = ldpair(&lp[rt * 16 + lo][kk + 2 * hi]);
        const v2f b = ldpair(&lkT[ct * 16 + lo][kk + 2 * hi]);
        accN[q] = wmma_f32(a, b, accN[q]);
      }
    }
    if (tid < 64) {
      float s = 0.0f;
      for (int j = 0; j < 64; ++j) s += lp[tid][j];
      den[tid] += s;
    }
  }
  __syncthreads();
  // DV^T = (NUM/den)^T  -> lkT  (written transposed for the next matmul)
#pragma unroll
  for (int q = 0; q < 2; ++q) {
    const int ti = 2 * w + q, rt = ti >> 2, ct = ti & 3;
#pragma unroll
    for (int v = 0; v < 8; ++v) {
      const int r = rt * 16 + v + 8 * hi;
      lkT[ct * 16 + lo][r] = accN[q][v] / den[r];
    }
  }
  __syncthreads();
#pragma unroll
  for (int i = 0; i < 16; ++i) {
    const int idx = tid + i * 256;
    lq[idx >> 6][idx & 63] = GammaInv[(size_t)bh * 4096 + idx];
  }
  __syncthreads();
  mm64T(lp, lq, lkT, w, lo, hi);  // M2 = GammaInv @ DV
  __syncthreads();
#pragma unroll
  for (int i = 0; i < 16; ++i) {
    const int idx = tid + i * 256;
    M2[(size_t)bh * 4096 + idx] = lp[idx >> 6][idx & 63];
  }
}

// K6: out = rownorm(exp(Q @ Qt^T / 8)) @ M2.  128-row strips, dynamic LDS.
__global__ __launch_bounds__(256) void final_kernel(
    const float* __restrict__ QKV, const float* __restrict__ Qt,
    const float* __restrict__ M2, float* __restrict__ out) {
  extern __shared__ float smem[];
  float(*lqs)[ST64] = (float(*)[ST64])smem;                          // 128 rows
  float(*lkt)[ST64] = (float(*)[ST64])(smem + 128 * ST64);           // Qt
  float(*lp)[ST64] = (float(*)[ST64])(smem + (128 + 64) * ST64);     // 128 rows
  float(*lm2t)[ST64] = (float(*)[ST64])(smem + (256 + 64) * ST64);   // M2^T
  float* den = smem + (256 + 128) * ST64;                            // 128
  const int bh = blockIdx.y;
  const int t0 = blockIdx.x * 128;
  const int tid = threadIdx.x;
  const int w = tid >> 5, lane = tid & 31, lo = lane & 15, hi = lane >> 4;

#pragma unroll
  for (int i = 0; i < 32; ++i) {
    const int idx = tid + i * 256;
    lqs[idx >> 6][idx & 63] = QKV[((size_t)bh * SEQ + t0) * HDIM + idx];
  }
#pragma unroll
  for (int i = 0; i < 16; ++i) {
    const int idx = tid + i * 256;
    const int l = idx >> 6, d = idx & 63;
    lkt[l][d] = Qt[(size_t)bh * 4096 + idx];
    lm2t[d][l] = M2[(size_t)bh * 4096 + idx];  // store transposed
  }
  __syncthreads();
  // P strip = exp(Q(16x64) @ Qt^T):  Bt = lkt
#pragma unroll
  for (int ct = 0; ct < 4; ++ct) {
    v8f acc = {};
#pragma unroll
    for (int kk = 0; kk < 64; kk += 4) {
      const v2f a = ldpair(&lqs[w * 16 + lo][kk + 2 * hi]);
      const v2f b = ldpair(&lkt[ct * 16 + lo][kk + 2 * hi]);
      acc = wmma_f32(a, b, acc);
    }
#pragma unroll
    for (int v = 0; v < 8; ++v)
      lp[w * 16 + v + 8 * hi][ct * 16 + lo] = __expf(acc[v] * 0.125f);
  }
  __syncthreads();
  if (tid < 128) {
    float s = 0.0f;
    for (int j = 0; j < 64; ++j) s += lp[tid][j];
    den[tid] = s;
  }
  __syncthreads();
  // out strip = (P / den) @ M2:  Bt = lm2t
#pragma unroll
  for (int ct = 0; ct < 4; ++ct) {
    v8f acc = {};
#pragma unroll
    for (int kk = 0; kk < 64; kk += 4) {
      const v2f a = ldpair(&lp[w * 16 + lo][kk + 2 * hi]);
      const v2f b = ldpair(&lm2t[ct * 16 + lo][kk + 2 * hi]);
      acc = wmma_f32(a, b, acc);
    }
#pragma unroll
    for (int v = 0; v < 8; ++v) {
      const int r = w * 16 + v + 8 * hi;
      out[((size_t)bh * SEQ + t0 + r) * HDIM + ct * 16 + lo] = acc[v] / den[r];
    }
  }
}

extern "C" void kernel_launch(void* const* d_in, const int* in_sizes, int n_in,
                              void* d_out, int out_size, void* d_ws,
                              size_t ws_size, hipStream_t stream) {
  const float* X = (const float*)d_in[0];   // [4, 1024, 4096]
  const float* Wq = (const float*)d_in[1];  // [16, 64, 1024]
  const float* bq = (const float*)d_in[2];  // [16, 64]
  float* out = (float*)d_out;               // [4, 16, 4096, 64]

  float* ws = (float*)d_ws;
  float* QKV = ws;                                       // 16,777,216 floats
  float* Qt = QKV + (size_t)BH * SEQ * HDIM;             // 262,144
  float* GammaD = Qt + (size_t)BH * NLMK * HDIM;         // 262,144
  float* GammaInv = GammaD + (size_t)BH * NLMK * NLMK;   // 262,144
  float* M2 = GammaInv + (size_t)BH * NLMK * NLMK;       // 262,144
  int* scaleMax = (int*)(M2 + (size_t)BH * NLMK * NLMK);

  init_kernel<<<1, 1, 0, stream>>>(scaleMax);
  proj_kernel<<<dim3(NHEAD, SEQ / 128, BATCH), 256, 0, stream>>>(X, Wq, bq, QKV);
  pool_kernel<<<(BH * NLMK * HDIM) / 256, 256, 0, stream>>>(QKV, Qt);
  gamma_kernel<<<BH, 256, 0, stream>>>(Qt, GammaD, scaleMax);
  newton_kernel<<<BH, 256, 5 * 64 * ST64 * sizeof(float), stream>>>(
      GammaD, scaleMax, GammaInv);
  deltav_kernel<<<BH, 256, (4 * 64 * ST64 + 64) * sizeof(float), stream>>>(
      QKV, Qt, GammaInv, M2);
  final_kernel<<<dim3(SEQ / 128, BH), 256,
                 ((256 + 128) * ST64 + 128) * sizeof(float), stream>>>(
      QKV, Qt, M2, out);
}